// GATv2Attention_46531675684919
// MI455X (gfx1250) — compile-verified
//
#include <hip/hip_runtime.h>
#include <cstdint>

// Problem constants (from reference): B,H,LQ,LK,D = 2,8,384,384,64
#define B_    2
#define H_    8
#define LQ_   384
#define LK_   384
#define D_    64

#define ROWS_WG      16                 // query rows per workgroup
#define ROWS_WAVE    2                  // query rows per wave (8 waves * 32 lanes = 256 threads)
#define TILES        (LQ_ / ROWS_WG)    // 24 row-tiles per (b,h)
#define CHUNK_J      192                // k rows staged in LDS per TDM transfer
#define KROW         68                 // padded LDS row stride in dwords (64 data + 4 pad)
#define JB_CHUNK     (CHUNK_J / 32)     // 6 j-blocks per chunk
#define NJB          (LK_ / 32)         // 12 j-blocks total

typedef unsigned int v4u __attribute__((ext_vector_type(4)));
typedef int          v8i __attribute__((ext_vector_type(8)));
typedef int          v4i __attribute__((ext_vector_type(4)));

// silu(x) = x * sigmoid(x) = h + h*tanh(h), h = x/2   -> 1 trans + 2 VALU
__device__ __forceinline__ float silu_fast(float x) {
  float h = 0.5f * x;
  float t;
#if __has_builtin(__builtin_amdgcn_tanhf)
  t = __builtin_amdgcn_tanhf(h);
#elif __has_builtin(__builtin_amdgcn_tanh_f32)
  t = __builtin_amdgcn_tanh_f32(h);
#else
  asm("v_tanh_f32 %0, %1" : "=v"(t) : "v"(h));
#endif
  return __builtin_fmaf(h, t, h);
}

// TDM: DMA a CHUNK_J x 64 f32 tile (row-major, contiguous) from global into LDS,
// inserting 4 pad dwords after every 64 data dwords -> LDS row stride = 68 dwords.
__device__ __forceinline__ void tdm_load_k_chunk(const float* gsrc, unsigned lds_byte_off) {
  unsigned long long ga = (unsigned long long)(size_t)gsrc;
  v4u g0;
  g0.x = 1u;                                   // [1:0] count=1 (valid user descriptor)
  g0.y = lds_byte_off;                         // [63:32] lds_addr (bytes)
  g0.z = (unsigned)(ga & 0xFFFFFFFFu);         // global_addr[31:0]
  g0.w = (unsigned)((ga >> 32) & 0x01FFFFFFu)  // global_addr[56:32]
       | (2u << 30);                           // [127:126] type = 2 ("image")
  v8i g1;
  g1[0] = (int)((2u << 16)                     // data_size = 4 bytes
              | (1u << 20)                     // pad_enable
              | (5u << 22)                     // pad_interval: every 64 dwords (256B)
              | (3u << 25));                   // pad_amount: 4 dwords
  g1[1] = (int)(((unsigned)D_) << 16);         // tensor_dim0[15:0] = 64 (in bits 63:48)
  g1[2] = (int)(((unsigned)CHUNK_J) << 16);    // tensor_dim1[15:0] = 192 (bits 95:80)
  g1[3] = (int)(((unsigned)D_) << 16);         // tile_dim0 = 64 (bits 127:112)
  g1[4] = (int)(CHUNK_J);                      // tile_dim1 = 192, tile_dim2 = 0
  g1[5] = (int)(D_);                           // tensor_dim0_stride = 64 elements
  g1[6] = 0;                                   // stride hi / tensor_dim1_stride lo (unused, 2D)
  g1[7] = 0;
  v4i z4 = {0, 0, 0, 0};                       // groups 2/3 unused (2D tensor)
  v8i z8 = {0, 0, 0, 0, 0, 0, 0, 0};           // extra group in 6-arg form: unused
  __builtin_amdgcn_tensor_load_to_lds(g0, g1, z4, z4, z8, 0);
}

__global__ __launch_bounds__(256, 1)
void gatv2_attn_kernel(const float* __restrict__ q,
                       const float* __restrict__ k,
                       const float* __restrict__ attn,
                       const unsigned char* __restrict__ mask,
                       float* __restrict__ out) {
  // Double-buffered k panel: both TDM transfers issued up front; chunk-1 DMA
  // overlaps chunk-0 compute. 2*52,224 + 4,096 + 256 = 108,800 B of LDS
  // (ISA allows up to 320 KB per workgroup on gfx1250).
  __shared__ __align__(16) float ksh[2][CHUNK_J * KROW];
  __shared__ __align__(16) float qsh[ROWS_WG * D_];
  __shared__ __align__(16) float ash[D_];

  const int tid  = threadIdx.x;
  const int bh   = blockIdx.x / TILES;
  const int tile = blockIdx.x % TILES;
  const int h    = bh % H_;
  const int i0   = tile * ROWS_WG;
  const int wave = tid >> 5;
  const int lane = tid & 31;
  const int r0   = wave * ROWS_WAVE;      // local q-row pair handled by this wave

  const float* kbase = k + (size_t)bh * LK_ * D_;

  // Kick off both DMA transfers first (in-order on TENSORcnt within wave 0).
  if (tid < 32) {
    tdm_load_k_chunk(kbase,                        (unsigned)(size_t)&ksh[0][0]);
    tdm_load_k_chunk(kbase + (size_t)CHUNK_J * D_, (unsigned)(size_t)&ksh[1][0]);
  }

  // Stage q tile (16*64 contiguous floats) and attention vector into LDS
  // while the TDM engine streams the k panel.
  {
    const float4* qg4 = (const float4*)(q + ((size_t)bh * LQ_ + i0) * D_);
    float4*       qs4 = (float4*)qsh;
    qs4[tid] = qg4[tid];                  // 256 float4 == 256 threads
    if (tid < D_) ash[tid] = attn[h * D_ + tid];
  }

  float acc0[NJB], acc1[NJB];
#pragma unroll
  for (int jb = 0; jb < NJB; ++jb) { acc0[jb] = 0.f; acc1[jb] = 0.f; }

  // acc index must be compile-time constant so accumulators stay in VGPRs.
#define ACCUM_CHUNK(C)                                                          \
  {                                                                             \
    _Pragma("unroll 2")                                                         \
    for (int dc = 0; dc < D_ / 4; ++dc) {                                       \
      const float4 av = *(const float4*)&ash[dc * 4];                           \
      const float4 q0 = *(const float4*)&qsh[r0 * D_ + dc * 4];                 \
      const float4 q1 = *(const float4*)&qsh[(r0 + 1) * D_ + dc * 4];           \
      _Pragma("unroll")                                                         \
      for (int jbl = 0; jbl < JB_CHUNK; ++jbl) {                                \
        const float4 kv = *(const float4*)&ksh[(C)][(jbl * 32 + lane) * KROW + dc * 4]; \
        float s0 = acc0[(C) * JB_CHUNK + jbl];                                  \
        float s1 = acc1[(C) * JB_CHUNK + jbl];                                  \
        s0 = __builtin_fmaf(silu_fast(q0.x + kv.x), av.x, s0);                  \
        s1 = __builtin_fmaf(silu_fast(q1.x + kv.x), av.x, s1);                  \
        s0 = __builtin_fmaf(silu_fast(q0.y + kv.y), av.y, s0);                  \
        s1 = __builtin_fmaf(silu_fast(q1.y + kv.y), av.y, s1);                  \
        s0 = __builtin_fmaf(silu_fast(q0.z + kv.z), av.z, s0);                  \
        s1 = __builtin_fmaf(silu_fast(q1.z + kv.z), av.z, s1);                  \
        s0 = __builtin_fmaf(silu_fast(q0.w + kv.w), av.w, s0);                  \
        s1 = __builtin_fmaf(silu_fast(q1.w + kv.w), av.w, s1);                  \
        acc0[(C) * JB_CHUNK + jbl] = s0;                                        \
        acc1[(C) * JB_CHUNK + jbl] = s1;                                        \
      }                                                                         \
    }                                                                           \
  }

  // chunk 0 ready when TENSORcnt <= 1 (tensor ops complete in order per wave)
  if (tid < 32) __builtin_amdgcn_s_wait_tensorcnt(1);
  __syncthreads();
  ACCUM_CHUNK(0)
  // chunk 1 ready when TENSORcnt == 0 (its DMA overlapped the compute above)
  if (tid < 32) __builtin_amdgcn_s_wait_tensorcnt(0);
  __syncthreads();
  ACCUM_CHUNK(1)

  // ---- masked softmax over Lk, fully inside one wave32 (shfl_xor reductions) ----
#define SOFTMAX_STORE(ACC, I)                                                   \
  {                                                                             \
    const size_t rowoff = ((size_t)bh * LQ_ + (size_t)(I)) * (size_t)LK_;       \
    const unsigned char* mrow = mask + rowoff;                                  \
    float e[NJB];                                                               \
    float mx = -__builtin_inff();                                               \
    _Pragma("unroll")                                                           \
    for (int jb = 0; jb < NJB; ++jb) {                                          \
      const unsigned m = mrow[jb * 32 + lane];                                  \
      const float s = m ? -__builtin_inff() : ACC[jb];                          \
      e[jb] = s;                                                                \
      mx = fmaxf(mx, s);                                                        \
    }                                                                           \
    _Pragma("unroll")                                                           \
    for (int off = 16; off > 0; off >>= 1) mx = fmaxf(mx, __shfl_xor(mx, off, 32)); \
    float sum = 0.f;                                                            \
    _Pragma("unroll")                                                           \
    for (int jb = 0; jb < NJB; ++jb) {                                          \
      const float ev = (e[jb] == -__builtin_inff()) ? 0.f : __expf(e[jb] - mx); \
      e[jb] = ev;                                                               \
      sum += ev;                                                                \
    }                                                                           \
    _Pragma("unroll")                                                           \
    for (int off = 16; off > 0; off >>= 1) sum += __shfl_xor(sum, off, 32);     \
    const float inv = (sum > 0.f) ? __builtin_amdgcn_rcpf(sum) : 0.f;           \
    _Pragma("unroll")                                                           \
    for (int jb = 0; jb < NJB; ++jb) out[rowoff + jb * 32 + lane] = e[jb] * inv; \
  }

  SOFTMAX_STORE(acc0, i0 + r0)
  SOFTMAX_STORE(acc1, i0 + r0 + 1)

#undef ACCUM_CHUNK
#undef SOFTMAX_STORE
}

extern "C" void kernel_launch(void* const* d_in, const int* in_sizes, int n_in,
                              void* d_out, int out_size, void* d_ws, size_t ws_size,
                              hipStream_t stream) {
  (void)in_sizes; (void)n_in; (void)out_size; (void)d_ws; (void)ws_size;
  const float*         q    = (const float*)d_in[0];
  const float*         k    = (const float*)d_in[1];
  const float*         attn = (const float*)d_in[2];  // (1,H,1,1,D)
  const unsigned char* mask = (const unsigned char*)d_in[3];  // jax bool = 1 byte/elt
  float*               out  = (float*)d_out;

  dim3 grid(B_ * H_ * TILES);  // 384 workgroups
  gatv2_attn_kernel<<<grid, 256, 0, stream>>>(q, k, attn, mask, out);
}